// FeatureSpecExtractor_29231547416748
// MI455X (gfx1250) — compile-verified
//
#include <hip/hip_runtime.h>
#include <hip/hip_bf16.h>

// ---------------- problem constants ----------------
#define NBATCH 16
#define NFEAT  256
#define NTIME  8000
#define SPEC_ROWS 257            // input has 257 feature rows; we use 256
#define ALPHA_F       0.95f
#define ONE_M_ALPHA_F 0.05f

// chunked-scan decomposition: 8000 = 25 chunks * 320, 320 = 20 tiles * 16
#define NCHUNK 25
#define TCHUNK 320
#define NTILE  20

typedef __attribute__((ext_vector_type(2))) float v2f;
typedef __attribute__((ext_vector_type(4))) float v4f;
typedef __attribute__((ext_vector_type(8))) float v8f;

// raw v_rsq_f32 (s is an EMA of |x| seeded >= 1e-4: never denormal, skip OCML guards)
__device__ __forceinline__ float fast_rsq(float x) {
  return __builtin_amdgcn_rsqf(x);
}

// ---------------------------------------------------------------------------
// Pass 1: per (b, f, chunk) compute B_c = sum_{j=0..319} (1-a) a^{319-j} |x_j|
// i.e. run the recurrence from s=0 through the chunk; keep only the end state.
// 102400 threads = 3200 waves -> saturates HBM read of ~131 MB.
// ---------------------------------------------------------------------------
__global__ void fse_pass1_chunk_contrib(const float* __restrict__ spec,
                                        float* __restrict__ Bc) {
  int tid = blockIdx.x * blockDim.x + threadIdx.x;
  if (tid >= NBATCH * NFEAT * NCHUNK) return;
  int c = tid % NCHUNK;
  int f = (tid / NCHUNK) % NFEAT;
  int b = tid / (NCHUNK * NFEAT);

  const float* p = spec + ((size_t)b * SPEC_ROWS + f) * NTIME + (size_t)c * TCHUNK;
  float s = 0.0f;
#pragma unroll 4
  for (int t = 0; t < TCHUNK; t += 4) {
    v4f x = *(const v4f*)(p + t);          // 16B-aligned stream
    s = fmaf(fabsf(x.x), ONE_M_ALPHA_F, s * ALPHA_F);
    s = fmaf(fabsf(x.y), ONE_M_ALPHA_F, s * ALPHA_F);
    s = fmaf(fabsf(x.z), ONE_M_ALPHA_F, s * ALPHA_F);
    s = fmaf(fabsf(x.w), ONE_M_ALPHA_F, s * ALPHA_F);
  }
  Bc[tid] = s;
}

// ---------------------------------------------------------------------------
// Pass 2: per (b, f) scan over 25 chunk boundaries:
//   S_store[c] = S;  S = a^320 * S + B_c
// ---------------------------------------------------------------------------
__global__ void fse_pass2_boundary_scan(const float* __restrict__ s0,
                                        const float* __restrict__ Bc,
                                        float* __restrict__ Sst) {
  int tid = blockIdx.x * blockDim.x + threadIdx.x;
  if (tid >= NBATCH * NFEAT) return;
  int f = tid % NFEAT;

  float decay = __powf(ALPHA_F, (float)TCHUNK);   // a^320 ~ 7.5e-8
  float S = s0[f];
  const float* bp = Bc + (size_t)tid * NCHUNK;
  float* sp = Sst + (size_t)tid * NCHUNK;
#pragma unroll
  for (int c = 0; c < NCHUNK; ++c) {
    sp[c] = S;
    S = fmaf(decay, S, bp[c]);
  }
}

// ---------------------------------------------------------------------------
// Pass 3: one wave per (batch, 16-feature group, chunk).
// Per 16-time x 16-feature tile:
//   S_tile = L * |X|  +  (alpha^{t+1}) (x) s_in      via 5 chained
//   V_WMMA_F32_16X16X4_F32 ops; then y = x * rsqrt(s) elementwise,
// carry = D row 15 broadcast by lane shuffle.
// ---------------------------------------------------------------------------
__global__ void fse_pass3_wmma_scan(const float* __restrict__ spec,
                                    const float* __restrict__ Sst,
                                    float* __restrict__ out) {
  int wave = (int)((blockIdx.x * blockDim.x + threadIdx.x) >> 5);
  int lane = threadIdx.x & 31;
  if (wave >= NBATCH * (NFEAT / 16) * NCHUNK) return;   // wave-uniform

  int c  = wave % NCHUNK;
  int fg = (wave / NCHUNK) % (NFEAT / 16);
  int b  = wave / (NCHUNK * (NFEAT / 16));

  int hl = lane >> 4;          // half-wave select (K-pair / row-group)
  int m  = lane & 15;          // A-row / B,C,D-column index
  int f  = fg * 16 + m;

  // powers of alpha up to 16 (per-lane registers)
  float pw[17];
  pw[0] = 1.0f;
#pragma unroll
  for (int i = 1; i <= 16; ++i) pw[i] = pw[i - 1] * ALPHA_F;

  // A-matrix slices of L[t,k] = (1-a) a^{t-k} (k<=t), K-slices of 4.
  // A 16x4 layout: lanes 0-15 hold (K0,K1) in (v0,v1); lanes 16-31 hold (K2,K3).
  v2f A[4];
#pragma unroll
  for (int j = 0; j < 4; ++j) {
    int k0 = 4 * j + 2 * hl;
    A[j].x = (k0     <= m) ? ONE_M_ALPHA_F * pw[m - k0]     : 0.0f;
    A[j].y = (k0 + 1 <= m) ? ONE_M_ALPHA_F * pw[m - k0 - 1] : 0.0f;
  }
  // carry A column: Ac[t,0] = a^{t+1}, cols 1..3 = 0
  v2f Ac;
  Ac.x = (hl == 0) ? pw[m + 1] : 0.0f;
  Ac.y = 0.0f;

  // chunk-entry state for this feature (valid/used from lanes 0..15)
  float s_in = Sst[((size_t)b * NFEAT + f) * NCHUNK + c];

  const float* xin  = spec + ((size_t)b * SPEC_ROWS + f) * NTIME + (size_t)c * TCHUNK;
  float*       yout = out  + ((size_t)b * NFEAT     + f) * NTIME + (size_t)c * TCHUNK;

  for (int tile = 0; tile < NTILE; ++tile) {
    const float* xt = xin + tile * 16;

    // B slices: B_j[k, f] = |x[t=4j+k, f]|, B 4x16 layout mirrors A layout.
    v2f Bv[4];
#pragma unroll
    for (int j = 0; j < 4; ++j) {
      v2f xv = *(const v2f*)(xt + 4 * j + 2 * hl);   // global_load_b64
      Bv[j].x = fabsf(xv.x);
      Bv[j].y = fabsf(xv.y);
    }
    // carry B row: Bc2[0, f] = s_in, rows 1..3 = 0
    v2f Bc2;
    Bc2.x = (hl == 0) ? s_in : 0.0f;
    Bc2.y = 0.0f;

    v8f d = {};
    d = __builtin_amdgcn_wmma_f32_16x16x4_f32(false, Ac,   false, Bc2,   (short)0, d, false, false);
    d = __builtin_amdgcn_wmma_f32_16x16x4_f32(false, A[0], false, Bv[0], (short)0, d, false, false);
    d = __builtin_amdgcn_wmma_f32_16x16x4_f32(false, A[1], false, Bv[1], (short)0, d, false, false);
    d = __builtin_amdgcn_wmma_f32_16x16x4_f32(false, A[2], false, Bv[2], (short)0, d, false, false);
    d = __builtin_amdgcn_wmma_f32_16x16x4_f32(false, A[3], false, Bv[3], (short)0, d, false, false);

    // next-tile carry: s at row 15 lives in lanes 16..31, VGPR 7, column m
    s_in = __shfl(d[7], 16 + m, 32);

    // elementwise y = x * rsqrt(s).  This lane covers t = tile*16 + 8*hl + v,
    // v = 0..7 -> two contiguous float4 loads / NT stores per lane.
    const float* xr = xt + 8 * hl;
    v4f x0 = *(const v4f*)(xr);
    v4f x1 = *(const v4f*)(xr + 4);
    v4f y0, y1;
    y0.x = x0.x * fast_rsq(d[0]);
    y0.y = x0.y * fast_rsq(d[1]);
    y0.z = x0.z * fast_rsq(d[2]);
    y0.w = x0.w * fast_rsq(d[3]);
    y1.x = x1.x * fast_rsq(d[4]);
    y1.y = x1.y * fast_rsq(d[5]);
    y1.z = x1.z * fast_rsq(d[6]);
    y1.w = x1.w * fast_rsq(d[7]);

    float* yw = yout + tile * 16 + 8 * hl;
    __builtin_nontemporal_store(y0, (v4f*)(yw));       // streamed output: keep L2 for input
    __builtin_nontemporal_store(y1, (v4f*)(yw + 4));
  }
}

// ---------------------------------------------------------------------------
extern "C" void kernel_launch(void* const* d_in, const int* in_sizes, int n_in,
                              void* d_out, int out_size, void* d_ws, size_t ws_size,
                              hipStream_t stream) {
  const float* spec = (const float*)d_in[0];      // (16,1,257,8000) f32
  const float* s0   = (const float*)d_in[1];      // (256,) f32
  float* out = (float*)d_out;                     // (16,1,256,8000) f32

  // workspace layout
  float* Bc  = (float*)d_ws;                                  // 16*256*25 floats
  float* Sst = Bc + (size_t)NBATCH * NFEAT * NCHUNK;          // 16*256*25 floats

  // Pass 1: 102400 threads
  {
    int total = NBATCH * NFEAT * NCHUNK;
    fse_pass1_chunk_contrib<<<(total + 255) / 256, 256, 0, stream>>>(spec, Bc);
  }
  // Pass 2: 4096 threads
  {
    int total = NBATCH * NFEAT;
    fse_pass2_boundary_scan<<<(total + 255) / 256, 256, 0, stream>>>(s0, Bc, Sst);
  }
  // Pass 3: 6400 waves = 204800 threads
  {
    int waves = NBATCH * (NFEAT / 16) * NCHUNK;
    int total = waves * 32;
    fse_pass3_wmma_scan<<<(total + 255) / 256, 256, 0, stream>>>(spec, Sst, out);
  }
}